// GPT_43946105372753
// MI455X (gfx1250) — compile-verified
//
#include <hip/hip_runtime.h>
#include <hip/hip_bf16.h>

// ---- problem constants (from reference) ----
#define Dm   512
#define Hh   8
#define Ll   6
#define Ss   1024
#define Bb   2
#define Vv   50257
#define DHd  64
#define MSr  (Bb * Ss)              // 2048 rows of activations
#define INV_SCALE 0.04419417382415922f   // 1/sqrt(512)

#define HEAD_COLS_TILED 50240       // 785 blocks * 64 cols; tail = 17 cols
#define HEAD_TAIL (Vv - HEAD_COLS_TILED)

typedef float v2f __attribute__((ext_vector_type(2)));
typedef float v8f __attribute__((ext_vector_type(8)));

// ---------------------------------------------------------------------------
// x[b,s,:] = (tok_emb[ids[b,s]] + pos_emb[s]) * mask[b,s]
// ---------------------------------------------------------------------------
__global__ void embed_kernel(const int* __restrict__ idx, const int* __restrict__ masks,
                             const float* __restrict__ tok, const float* __restrict__ pos,
                             float* __restrict__ x) {
    int i = blockIdx.x * blockDim.x + threadIdx.x;
    if (i >= MSr * Dm) return;
    int row = i / Dm, d = i - row * Dm;
    int b = row / Ss, s = row - b * Ss;
    int id = idx[b * (Ss + 1) + s];
    float m = (float)masks[row];
    x[i] = (tok[(size_t)id * Dm + d] + pos[s * Dm + d]) * m;
}

// ---------------------------------------------------------------------------
// LayerNorm: one wave32 per row of D=512 (16 elements per lane), shfl butterfly
// ---------------------------------------------------------------------------
__global__ void ln_kernel(const float* __restrict__ x, const float* __restrict__ g,
                          const float* __restrict__ b, float* __restrict__ y) {
    int row  = blockIdx.x;
    int lane = threadIdx.x;
    const float* xr = x + (size_t)row * Dm;
    float v[16];
    float s = 0.f;
#pragma unroll
    for (int i = 0; i < 16; ++i) { v[i] = xr[lane + i * 32]; s += v[i]; }
#pragma unroll
    for (int o = 16; o >= 1; o >>= 1) s += __shfl_xor(s, o, 32);
    float mean = s * (1.0f / Dm);
    float vs = 0.f;
#pragma unroll
    for (int i = 0; i < 16; ++i) { float t = v[i] - mean; vs += t * t; }
#pragma unroll
    for (int o = 16; o >= 1; o >>= 1) vs += __shfl_xor(vs, o, 32);
    float inv = rsqrtf(vs * (1.0f / Dm) + 1e-5f);
    float* yr = y + (size_t)row * Dm;
#pragma unroll
    for (int i = 0; i < 16; ++i) {
        int d = lane + i * 32;
        yr[d] = (v[i] - mean) * inv * g[d] + b[d];
    }
}

// ---------------------------------------------------------------------------
// fp32 GEMM on the matrix pipes: C = act(A @ W + bias) (+ residual)
// One wave computes a 16 x (16*NT) tile of C. All dims compile-time, no guards:
// K-loop in steps of 4, A fragment (v2f, 8B load) reused across NT wmmas.
// Fragment layout per ISA 7.12.2 (32-bit A 16x4 / B 4x16 / C 16x16).
// ---------------------------------------------------------------------------
template<int K, int N, int NT, bool GELU, bool BIAS, bool RES>
__global__ void gemm_wmma_kernel(const float* __restrict__ A, const float* __restrict__ W,
                                 const float* __restrict__ bias, const float* __restrict__ res,
                                 float* __restrict__ C) {
    const int lane = threadIdx.x;
    const int rlo  = lane & 15;
    const int hi   = lane >> 4;                 // 0 or 1
    const int row0 = blockIdx.y * 16;
    const int col0 = blockIdx.x * (16 * NT);

    const float* arow = A + (size_t)(row0 + rlo) * K;              // lane l: row l%16 of A
    const float* wp   = W + (size_t)(2 * hi) * N + col0 + rlo;     // rows k0+2hi, k0+2hi+1

    v8f acc[NT];
#pragma unroll
    for (int t = 0; t < NT; ++t) acc[t] = (v8f){};

    for (int k0 = 0; k0 < K; k0 += 4) {
        // A frag: a[j] = A[row0+rlo][k0 + 2*hi + j]
        v2f a = *(const v2f*)(arow + k0 + hi * 2);
#pragma unroll
        for (int t = 0; t < NT; ++t) {
            v2f b;                              // b[j] = W[k0+2hi+j][col0 + 16t + rlo]
            b[0] = wp[t * 16];
            b[1] = wp[t * 16 + N];
            acc[t] = __builtin_amdgcn_wmma_f32_16x16x4_f32(false, a, false, b,
                                                           (short)0, acc[t], false, false);
        }
        wp += (size_t)4 * N;
    }

#pragma unroll
    for (int t = 0; t < NT; ++t) {
        const int col = col0 + t * 16 + rlo;
        const float bv = BIAS ? bias[col] : 0.f;
#pragma unroll
        for (int r = 0; r < 8; ++r) {
            const size_t off = (size_t)(row0 + hi * 8 + r) * N + col;
            float val = acc[t][r] + bv;
            if (GELU) val = 0.5f * val * (1.f + erff(val * 0.70710678118654752f));
            if (RES)  val += res[off];
            C[off] = val;
        }
    }
}

// Ragged tail of the LM head: 17 columns x 2048 rows, plain dot products.
__global__ void head_tail_kernel(const float* __restrict__ A, const float* __restrict__ W,
                                 float* __restrict__ C) {
    int i = blockIdx.x * blockDim.x + threadIdx.x;
    if (i >= MSr * HEAD_TAIL) return;
    int row = i / HEAD_TAIL, c = i - row * HEAD_TAIL;
    int col = HEAD_COLS_TILED + c;
    const float* ar = A + (size_t)row * Dm;
    float s = 0.f;
#pragma unroll 8
    for (int k = 0; k < Dm; ++k) s += ar[k] * W[(size_t)k * Vv + col];
    C[(size_t)row * Vv + col] = s;
}

// ---------------------------------------------------------------------------
// Causal attention, online softmax. One wave32 per (b, h, q); DH=64 => 2/lane.
// Reference scales scores by 1/sqrt(D) = 1/sqrt(512).
// ---------------------------------------------------------------------------
__global__ void attn_kernel(const float* __restrict__ Q, const float* __restrict__ Kt,
                            const float* __restrict__ Vt, float* __restrict__ O) {
    int id   = blockIdx.x;            // b*H*S + h*S + q
    int q    = id % Ss;
    int bh   = id / Ss;
    int h    = bh % Hh;
    int b    = bh / Hh;
    int lane = threadIdx.x;
    size_t base = (size_t)b * Ss;

    const float* qp = Q + (base + q) * Dm + h * DHd;
    float q0 = qp[lane * 2], q1 = qp[lane * 2 + 1];

    float m = -INFINITY, l = 0.f, a0 = 0.f, a1 = 0.f;
    for (int j = 0; j <= q; ++j) {
        const float* kp = Kt + (base + j) * Dm + h * DHd;
        float d = q0 * kp[lane * 2] + q1 * kp[lane * 2 + 1];
#pragma unroll
        for (int o = 16; o >= 1; o >>= 1) d += __shfl_xor(d, o, 32);
        d *= INV_SCALE;
        float mn   = fmaxf(m, d);
        float corr = expf(m - mn);     // first iter: exp(-inf)=0
        float p    = expf(d - mn);
        const float* vp = Vt + (base + j) * Dm + h * DHd;
        l  = l  * corr + p;
        a0 = a0 * corr + p * vp[lane * 2];
        a1 = a1 * corr + p * vp[lane * 2 + 1];
        m = mn;
    }
    float inv = 1.f / l;
    float* op = O + (base + q) * Dm + h * DHd;
    op[lane * 2]     = a0 * inv;
    op[lane * 2 + 1] = a1 * inv;
}

// ---------------------------------------------------------------------------
// Per-row NLL: rowloss[row] = (logsumexp(logits[row]) - logits[row][target]) * mask
// ---------------------------------------------------------------------------
__global__ void rowloss_kernel(const float* __restrict__ logits, const int* __restrict__ idx,
                               const int* __restrict__ masks, float* __restrict__ rowloss) {
    int row = blockIdx.x;
    int b = row / Ss, s = row - b * Ss;
    int tid = threadIdx.x;
    const float* lr = logits + (size_t)row * Vv;
    __shared__ float red[256];

    float mx = -INFINITY;
    for (int j = tid; j < Vv; j += 256) mx = fmaxf(mx, lr[j]);
    red[tid] = mx; __syncthreads();
    for (int o = 128; o >= 1; o >>= 1) {
        if (tid < o) red[tid] = fmaxf(red[tid], red[tid + o]);
        __syncthreads();
    }
    mx = red[0]; __syncthreads();

    float se = 0.f;
    for (int j = tid; j < Vv; j += 256) se += expf(lr[j] - mx);
    red[tid] = se; __syncthreads();
    for (int o = 128; o >= 1; o >>= 1) {
        if (tid < o) red[tid] += red[tid + o];
        __syncthreads();
    }
    if (tid == 0) {
        float lse = mx + logf(red[0]);
        int t = idx[b * (Ss + 1) + s + 1];        // targets = idx[:,1:]
        rowloss[row] = (lse - lr[t]) * (float)masks[row];
    }
}

// Deterministic fixed-order reduction of 2048 row losses -> loss = sum / (B*S)
__global__ void loss_reduce_kernel(const float* __restrict__ rowloss, float* __restrict__ out) {
    __shared__ float red[256];
    int tid = threadIdx.x;
    float s = 0.f;
    for (int j = tid; j < MSr; j += 256) s += rowloss[j];
    red[tid] = s; __syncthreads();
    for (int o = 128; o >= 1; o >>= 1) {
        if (tid < o) red[tid] += red[tid + o];
        __syncthreads();
    }
    if (tid == 0) out[(size_t)MSr * Vv] = red[0] / (float)MSr;
}

// ---------------------------------------------------------------------------
extern "C" void kernel_launch(void* const* d_in, const int* in_sizes, int n_in,
                              void* d_out, int out_size, void* d_ws, size_t ws_size,
                              hipStream_t stream) {
    const int*   idx   = (const int*)  d_in[0];
    const int*   masks = (const int*)  d_in[1];
    const float* tok   = (const float*)d_in[2];
    const float* pos   = (const float*)d_in[3];
    const float* Wq    = (const float*)d_in[4];
    const float* bq    = (const float*)d_in[5];
    const float* Wk    = (const float*)d_in[6];
    const float* bk    = (const float*)d_in[7];
    const float* Wv    = (const float*)d_in[8];
    const float* bv    = (const float*)d_in[9];
    const float* Wo    = (const float*)d_in[10];
    const float* bo    = (const float*)d_in[11];
    const float* ln1g  = (const float*)d_in[12];
    const float* ln1b  = (const float*)d_in[13];
    const float* W1    = (const float*)d_in[14];
    const float* b1    = (const float*)d_in[15];
    const float* W2    = (const float*)d_in[16];
    const float* b2    = (const float*)d_in[17];
    const float* ln2g  = (const float*)d_in[18];
    const float* ln2b  = (const float*)d_in[19];
    const float* lnfg  = (const float*)d_in[20];
    const float* lnfb  = (const float*)d_in[21];
    const float* headw = (const float*)d_in[22];

    const size_t MSD = (size_t)MSr * Dm;          // 1,048,576 floats
    float* ws = (float*)d_ws;
    float* X   = ws;                               // activations (residual stream)
    float* Hb  = ws + 1 * MSD;                     // LN output
    float* Qb  = ws + 2 * MSD;
    float* Kb  = ws + 3 * MSD;
    float* Vb  = ws + 4 * MSD;
    float* Ab  = ws + 5 * MSD;                     // attention output
    float* Fb  = ws + 6 * MSD;                     // FFN hidden (MSr x 4D)
    float* RL  = Fb + (size_t)MSr * 4 * Dm;        // 2048 row losses

    float* logits = (float*)d_out;                 // [MSr * Vv] then loss at the end

    // embeddings
    embed_kernel<<<(MSr * Dm + 255) / 256, 256, 0, stream>>>(idx, masks, tok, pos, X);

    dim3 gDD(Dm / 64, MSr / 16);                   // 8  x 128 : 16x64 tiles
    dim3 gD4(4 * Dm / 64, MSr / 16);               // 32 x 128

    for (int l = 0; l < Ll; ++l) {
        const size_t wDD = (size_t)l * Dm * Dm;
        const size_t wD4 = (size_t)l * Dm * 4 * Dm;

        ln_kernel<<<MSr, 32, 0, stream>>>(X, ln1g + l * Dm, ln1b + l * Dm, Hb);
        gemm_wmma_kernel<Dm, Dm, 4, false, true, false>
            <<<gDD, 32, 0, stream>>>(Hb, Wq + wDD, bq + l * Dm, nullptr, Qb);
        gemm_wmma_kernel<Dm, Dm, 4, false, true, false>
            <<<gDD, 32, 0, stream>>>(Hb, Wk + wDD, bk + l * Dm, nullptr, Kb);
        gemm_wmma_kernel<Dm, Dm, 4, false, true, false>
            <<<gDD, 32, 0, stream>>>(Hb, Wv + wDD, bv + l * Dm, nullptr, Vb);
        attn_kernel<<<Bb * Hh * Ss, 32, 0, stream>>>(Qb, Kb, Vb, Ab);
        gemm_wmma_kernel<Dm, Dm, 4, false, true, true>
            <<<gDD, 32, 0, stream>>>(Ab, Wo + wDD, bo + l * Dm, X, X);

        ln_kernel<<<MSr, 32, 0, stream>>>(X, ln2g + l * Dm, ln2b + l * Dm, Hb);
        gemm_wmma_kernel<Dm, 4 * Dm, 4, true, true, false>
            <<<gD4, 32, 0, stream>>>(Hb, W1 + wD4, b1 + (size_t)l * 4 * Dm, nullptr, Fb);
        gemm_wmma_kernel<4 * Dm, Dm, 4, false, true, true>
            <<<gDD, 32, 0, stream>>>(Fb, W2 + wD4, b2 + l * Dm, X, X);
    }

    // final LN + LM head (memory-bound: 412 MB logits)
    ln_kernel<<<MSr, 32, 0, stream>>>(X, lnfg, lnfb, Hb);
    dim3 gV(HEAD_COLS_TILED / 64, MSr / 16);       // 785 x 128 clean 16x64 tiles
    gemm_wmma_kernel<Dm, Vv, 4, false, false, false>
        <<<gV, 32, 0, stream>>>(Hb, headw, nullptr, nullptr, logits);
    head_tail_kernel<<<(MSr * HEAD_TAIL + 255) / 256, 256, 0, stream>>>(Hb, headw, logits);

    // loss
    rowloss_kernel<<<MSr, 256, 0, stream>>>(logits, idx, masks, RL);
    loss_reduce_kernel<<<1, 256, 0, stream>>>(RL, logits);
}